// Net7_29755533427165
// MI455X (gfx1250) — compile-verified
//
#include <hip/hip_runtime.h>
#include <hip/hip_bf16.h>

#define N_NODES 100000
#define N_EDGES 3200000
#define D_IN 16
#define D_OUT 32
#define N_TILES (N_NODES / 16)   // 6250, exact

typedef float v2f __attribute__((ext_vector_type(2)));
typedef float v8f __attribute__((ext_vector_type(8)));

// ---------------------------------------------------------------------------
// Zero the aggregation buffers (sum[N*16] followed contiguously by cnt[N]).
// ---------------------------------------------------------------------------
__global__ __launch_bounds__(256) void zero_kernel(float* __restrict__ p, int count) {
    int i = blockIdx.x * 256 + threadIdx.x;
    if (i < count) p[i] = 0.0f;
}

// ---------------------------------------------------------------------------
// h = relu(x @ w1 + b1) @ w2 + b2      (one wave per 16-node tile)
// Chain of 4x v_wmma_f32_16x16x4_f32 per GEMM (K=16).
// ---------------------------------------------------------------------------
__global__ __launch_bounds__(256) void mlp_kernel(
    const float* __restrict__ x,
    const float* __restrict__ w1, const float* __restrict__ b1,
    const float* __restrict__ w2, const float* __restrict__ b2,
    float* __restrict__ h)
{
    __shared__ float lds[8][16][17];           // per-wave transpose bounce
    const int lane = threadIdx.x & 31;
    const int wave = threadIdx.x >> 5;
    const int tile = blockIdx.x * 8 + wave;
    if (tile >= N_TILES) return;               // uniform per-wave: EXEC stays full
    const int hf = lane >> 4;                  // lane half (0/1)
    const int n  = lane & 15;                  // row (A) / col (B,C) index
    const long row = (long)tile * 16 + n;

    // ---- GEMM 1: X(16x16) @ W1(16x16) ----
    v2f a[4];
#pragma unroll
    for (int i = 0; i < 4; ++i) {
        const float* p = x + row * D_IN + 4 * i + 2 * hf;
        a[i] = (v2f){ p[0], p[1] };
    }
    v8f c = {};
#pragma unroll
    for (int i = 0; i < 4; ++i) {
        v2f b;
        b.x = w1[(4 * i + 2 * hf)     * D_IN + n];
        b.y = w1[(4 * i + 2 * hf + 1) * D_IN + n];
        c = __builtin_amdgcn_wmma_f32_16x16x4_f32(false, a[i], false, b,
                                                  (short)0, c, false, false);
    }
    // bias + ReLU, then re-lay-out (C layout -> A layout) through LDS
    const float bias1 = b1[n];
#pragma unroll
    for (int j = 0; j < 8; ++j) {
        float v = c[j] + bias1;
        lds[wave][j + 8 * hf][n] = v > 0.0f ? v : 0.0f;   // element (M=j+8*hf, N=n)
    }
    // same-wave DS store->load is in-order; compiler inserts s_wait_dscnt
    v2f a2[4];
#pragma unroll
    for (int i = 0; i < 4; ++i) {
        a2[i].x = lds[wave][n][4 * i + 2 * hf];
        a2[i].y = lds[wave][n][4 * i + 2 * hf + 1];
    }

    // ---- GEMM 2: relu(H1) @ W2 ----
    v8f c2 = {};
#pragma unroll
    for (int i = 0; i < 4; ++i) {
        v2f b;
        b.x = w2[(4 * i + 2 * hf)     * D_IN + n];
        b.y = w2[(4 * i + 2 * hf + 1) * D_IN + n];
        c2 = __builtin_amdgcn_wmma_f32_16x16x4_f32(false, a2[i], false, b,
                                                   (short)0, c2, false, false);
    }
    const float bias2 = b2[n];
#pragma unroll
    for (int j = 0; j < 8; ++j)
        h[((long)tile * 16 + j + 8 * hf) * D_IN + n] = c2[j] + bias2;
}

// ---------------------------------------------------------------------------
// Edge scatter: sum[dst] += h[src]; cnt[dst] += 1.   16 lanes per edge.
// sum/cnt (~7 MB) stay resident in the 192 MB L2 -> atomics resolve at L2.
// ---------------------------------------------------------------------------
__global__ __launch_bounds__(256) void scatter_kernel(
    const int* __restrict__ ei,          // [2, N_EDGES]
    const float* __restrict__ h,
    float* __restrict__ sum, float* __restrict__ cnt)
{
    const int t = blockIdx.x * 256 + threadIdx.x;
    const int e = t >> 4;
    const int f = t & 15;
    if (e >= N_EDGES) return;
    const int src = ei[e];
    const int dst = ei[N_EDGES + e];
    atomicAdd(&sum[(long)dst * D_IN + f], h[(long)src * D_IN + f]);
    if (f == 0) atomicAdd(&cnt[dst], 1.0f);
}

// ---------------------------------------------------------------------------
// out = (sum/max(cnt,1)) @ wl + bl + h @ wr   (one wave per 16-node tile,
// two 16-column halves of the 32-wide output, 8 WMMAs each)
// ---------------------------------------------------------------------------
__global__ __launch_bounds__(256) void sage_kernel(
    const float* __restrict__ h,
    const float* __restrict__ sum, const float* __restrict__ cnt,
    const float* __restrict__ wl, const float* __restrict__ bl,
    const float* __restrict__ wr,
    float* __restrict__ out)
{
    const int lane = threadIdx.x & 31;
    const int wave = threadIdx.x >> 5;
    const int tile = blockIdx.x * 8 + wave;
    if (tile >= N_TILES) return;
    const int hf = lane >> 4;
    const int n  = lane & 15;
    const long row = (long)tile * 16 + n;

    const float inv = 1.0f / fmaxf(cnt[row], 1.0f);

    v2f am[4], ah[4];
#pragma unroll
    for (int i = 0; i < 4; ++i) {
        const float* ps = sum + row * D_IN + 4 * i + 2 * hf;
        am[i] = (v2f){ ps[0] * inv, ps[1] * inv };
        const float* ph = h + row * D_IN + 4 * i + 2 * hf;
        ah[i] = (v2f){ ph[0], ph[1] };
    }

#pragma unroll
    for (int nh = 0; nh < 2; ++nh) {
        const float bias = bl[nh * 16 + n];
        v8f c;
#pragma unroll
        for (int j = 0; j < 8; ++j) c[j] = bias;
#pragma unroll
        for (int i = 0; i < 4; ++i) {             // mean @ wl
            v2f b;
            b.x = wl[(4 * i + 2 * hf)     * D_OUT + nh * 16 + n];
            b.y = wl[(4 * i + 2 * hf + 1) * D_OUT + nh * 16 + n];
            c = __builtin_amdgcn_wmma_f32_16x16x4_f32(false, am[i], false, b,
                                                      (short)0, c, false, false);
        }
#pragma unroll
        for (int i = 0; i < 4; ++i) {             // h @ wr
            v2f b;
            b.x = wr[(4 * i + 2 * hf)     * D_OUT + nh * 16 + n];
            b.y = wr[(4 * i + 2 * hf + 1) * D_OUT + nh * 16 + n];
            c = __builtin_amdgcn_wmma_f32_16x16x4_f32(false, ah[i], false, b,
                                                      (short)0, c, false, false);
        }
#pragma unroll
        for (int j = 0; j < 8; ++j)
            out[((long)tile * 16 + j + 8 * hf) * D_OUT + nh * 16 + n] = c[j];
    }
}

// ---------------------------------------------------------------------------
extern "C" void kernel_launch(void* const* d_in, const int* in_sizes, int n_in,
                              void* d_out, int out_size, void* d_ws, size_t ws_size,
                              hipStream_t stream) {
    const float* x  = (const float*)d_in[0];
    const int*   ei = (const int*)  d_in[1];
    const float* w1 = (const float*)d_in[2];
    const float* b1 = (const float*)d_in[3];
    const float* w2 = (const float*)d_in[4];
    const float* b2 = (const float*)d_in[5];
    const float* wl = (const float*)d_in[6];
    const float* bl = (const float*)d_in[7];
    const float* wr = (const float*)d_in[8];
    float* out = (float*)d_out;

    float* ws  = (float*)d_ws;
    float* h   = ws;                                // N*16 floats
    float* sum = ws + (size_t)N_NODES * 16;         // N*16 floats
    float* cnt = ws + (size_t)N_NODES * 32;         // N   floats (contiguous after sum)

    // 1) zero sum+cnt (contiguous N*17 floats)
    zero_kernel<<<(N_NODES * 17 + 255) / 256, 256, 0, stream>>>(sum, N_NODES * 17);
    // 2) per-node MLP via WMMA
    mlp_kernel<<<(N_TILES + 7) / 8, 256, 0, stream>>>(x, w1, b1, w2, b2, h);
    // 3) edge scatter (16 lanes/edge)
    scatter_kernel<<<(N_EDGES * 16) / 256, 256, 0, stream>>>(ei, h, sum, cnt);
    // 4) SAGE output via WMMA
    sage_kernel<<<(N_TILES + 7) / 8, 256, 0, stream>>>(h, sum, cnt, wl, bl, wr, out);
}